// BasicGNNConv_30700426232196
// MI455X (gfx1250) — compile-verified
//
#include <hip/hip_runtime.h>

// ---------------------------------------------------------------------------
// BasicGNNConv on MI455X (gfx1250), fp32 end-to-end via V_WMMA_F32_16X16X4_F32.
// Bandwidth-bound op (≈0.5 GB HBM traffic vs 26 GFLOP) -> keep fp32 precision.
// Round 2: weights pre-transposed so each B fragment is ONE global_load_b64
// (was 2x global_load_b32), halving inner-loop VMEM instruction count.
// ---------------------------------------------------------------------------

typedef __attribute__((ext_vector_type(2))) float v2f;
typedef __attribute__((ext_vector_type(8))) float v8f;

#define N_NODES  40000
#define N_EDGES  640000
#define NODE_DIM 256
#define EDGE_DIM 128
#define OUT_DIM  128
#define WAVES_PER_BLOCK 4

__device__ __forceinline__ v8f wmma_f32(v2f a, v2f b, v8f c) {
  // D = A(16x4) x B(4x16) + C(16x16), fp32, wave32
  return __builtin_amdgcn_wmma_f32_16x16x4_f32(
      /*neg_a=*/false, a, /*neg_b=*/false, b,
      /*c_mod=*/(short)0, c, /*reuse_a=*/false, /*reuse_b=*/false);
}

// ---------------------------------------------------------------------------
// Weight transpose: WT[n][k] = W[k][n].  (K x Nout -> Nout x K)
// Weights are <=128KB; runs once per launch, stays L2-resident afterwards.
// ---------------------------------------------------------------------------
__global__ __launch_bounds__(256) void transpose_kernel(
    const float* __restrict__ W, float* __restrict__ WT, int K, int Nout) {
  const int idx = blockIdx.x * blockDim.x + threadIdx.x;
  if (idx < K * Nout) {
    const int k = idx / Nout;
    const int n = idx % Nout;
    WT[(size_t)n * K + k] = W[idx];
  }
}

// ---------------------------------------------------------------------------
// Kernel 1: h = node_feat @ Wn + bn      [40000,256] x [256,128] -> [40000,128]
// One wave computes a 16(M) x 128(N) tile: 8 v8f accumulators.
// ---------------------------------------------------------------------------
__global__ __launch_bounds__(128) void gemm_h_kernel(
    const float* __restrict__ X, const float* __restrict__ WT,  // WT: [128][256]
    const float* __restrict__ bias, float* __restrict__ Y) {
  const int wave = threadIdx.x >> 5;
  const int lane = threadIdx.x & 31;
  const int ln   = lane & 15;
  const int half = lane >> 4;

  const int  tile  = blockIdx.x * WAVES_PER_BLOCK + wave;   // 2500 tiles exactly
  const int  row_a = tile * 16 + ln;                        // A-row owned by this lane
  const float* xrow  = X  + (size_t)row_a * NODE_DIM;
  const float* wbase = WT + (size_t)ln * NODE_DIM + 2 * half; // col n=ln (+16/tile)

  const v8f vzero = {0.f, 0.f, 0.f, 0.f, 0.f, 0.f, 0.f, 0.f};
  v8f acc[8];
#pragma unroll
  for (int t = 0; t < 8; ++t) acc[t] = vzero;

  for (int k0 = 0; k0 < NODE_DIM; k0 += 4) {
    // A fragment: lanes 0-15 carry K = k0+0,k0+1 ; lanes 16-31 carry K = k0+2,k0+3
    v2f a = *(const v2f*)(xrow + k0 + 2 * half);
#pragma unroll
    for (int t = 0; t < 8; ++t) {
      // B fragment: WT[t*16+ln][k0+2*half], WT[t*16+ln][k0+2*half+1] -> one b64
      v2f b = *(const v2f*)(wbase + (size_t)t * 16 * NODE_DIM + k0);
      acc[t] = wmma_f32(a, b, acc[t]);
    }
  }

  const int out_row0 = tile * 16;
#pragma unroll
  for (int t = 0; t < 8; ++t) {
    const int n = t * 16 + ln;
    const float bv = bias[n];
#pragma unroll
    for (int j = 0; j < 8; ++j) {
      // C/D layout: VGPR j -> M = j + 8*half, N = ln
      Y[(size_t)(out_row0 + j + 8 * half) * OUT_DIM + n] = acc[t][j] + bv;
    }
  }
}

// ---------------------------------------------------------------------------
// Kernel 2: degree count  deg[dst[i]] += 1.0
// ---------------------------------------------------------------------------
__global__ __launch_bounds__(256) void deg_kernel(
    const int* __restrict__ dst, float* __restrict__ deg) {
  const int i = blockIdx.x * blockDim.x + threadIdx.x;
  if (i < N_EDGES) atomicAdd(&deg[dst[i]], 1.0f);
}

// ---------------------------------------------------------------------------
// Kernel 3: m = edge_feat @ We + be + h[src]; agg_sum[dst] += m  (fused; the
// E x 128 intermediate never touches memory). One wave: 16 edges x 128 outputs.
// ---------------------------------------------------------------------------
__global__ __launch_bounds__(128) void edge_gemm_scatter_kernel(
    const float* __restrict__ EF, const float* __restrict__ WT, // WT: [128][128]
    const float* __restrict__ bias, const float* __restrict__ H,
    const int* __restrict__ src, const int* __restrict__ dst,
    float* __restrict__ aggsum) {
  const int wave = threadIdx.x >> 5;
  const int lane = threadIdx.x & 31;
  const int ln   = lane & 15;
  const int half = lane >> 4;

  const int  tile  = blockIdx.x * WAVES_PER_BLOCK + wave;   // 40000 tiles exactly
  const int  row_a = tile * 16 + ln;
  const float* xrow  = EF + (size_t)row_a * EDGE_DIM;
  const float* wbase = WT + (size_t)ln * EDGE_DIM + 2 * half;

  const v8f vzero = {0.f, 0.f, 0.f, 0.f, 0.f, 0.f, 0.f, 0.f};
  v8f acc[8];
#pragma unroll
  for (int t = 0; t < 8; ++t) acc[t] = vzero;

  for (int k0 = 0; k0 < EDGE_DIM; k0 += 4) {
    v2f a = *(const v2f*)(xrow + k0 + 2 * half);
#pragma unroll
    for (int t = 0; t < 8; ++t) {
      v2f b = *(const v2f*)(wbase + (size_t)t * 16 * EDGE_DIM + k0);
      acc[t] = wmma_f32(a, b, acc[t]);
    }
  }

  // Each lane owns output rows (edges) e0 + j + 8*half for j = 0..7.
  const int e0 = tile * 16;
  int sidx[8], didx[8];
#pragma unroll
  for (int j = 0; j < 8; ++j) {
    const int eid = e0 + j + 8 * half;
    sidx[j] = src[eid];
    didx[j] = dst[eid];
  }

#pragma unroll
  for (int t = 0; t < 8; ++t) {
    const int n = t * 16 + ln;
    const float bv = bias[n];
#pragma unroll
    for (int j = 0; j < 8; ++j) {
      const float v = acc[t][j] + bv + H[(size_t)sidx[j] * OUT_DIM + n];
      atomicAdd(&aggsum[(size_t)didx[j] * OUT_DIM + n], v);
    }
  }
}

// ---------------------------------------------------------------------------
// Kernel 4: out = concat(h, agg_sum/max(deg,1)) @ Wc + bc
// Same GEMM skeleton, K = 256; second K-half streams scaled agg rows.
// ---------------------------------------------------------------------------
__global__ __launch_bounds__(128) void combine_kernel(
    const float* __restrict__ H, const float* __restrict__ AGG,
    const float* __restrict__ deg, const float* __restrict__ WT, // WT: [128][256]
    const float* __restrict__ bias, float* __restrict__ OUT) {
  const int wave = threadIdx.x >> 5;
  const int lane = threadIdx.x & 31;
  const int ln   = lane & 15;
  const int half = lane >> 4;

  const int  tile  = blockIdx.x * WAVES_PER_BLOCK + wave;   // 2500 tiles exactly
  const int  row_a = tile * 16 + ln;
  const float* hrow  = H   + (size_t)row_a * OUT_DIM;
  const float* arow  = AGG + (size_t)row_a * OUT_DIM;
  const float* wbase = WT + (size_t)ln * (2 * OUT_DIM) + 2 * half;
  const float scale = 1.0f / fmaxf(deg[row_a], 1.0f);  // per-lane scalar (one A-row/lane)

  const v8f vzero = {0.f, 0.f, 0.f, 0.f, 0.f, 0.f, 0.f, 0.f};
  v8f acc[8];
#pragma unroll
  for (int t = 0; t < 8; ++t) acc[t] = vzero;

  for (int k0 = 0; k0 < 2 * OUT_DIM; k0 += 4) {
    const int k = k0 + 2 * half;          // stays within the same 128-half as k0
    v2f a;
    if (k0 < OUT_DIM) {                   // uniform branch across the wave
      a = *(const v2f*)(hrow + k);
    } else {
      v2f t2 = *(const v2f*)(arow + (k - OUT_DIM));
      a.x = t2.x * scale;
      a.y = t2.y * scale;
    }
#pragma unroll
    for (int t = 0; t < 8; ++t) {
      v2f b = *(const v2f*)(wbase + (size_t)t * 16 * (2 * OUT_DIM) + k0);
      acc[t] = wmma_f32(a, b, acc[t]);
    }
  }

  const int out_row0 = tile * 16;
#pragma unroll
  for (int t = 0; t < 8; ++t) {
    const int n = t * 16 + ln;
    const float bv = bias[n];
#pragma unroll
    for (int j = 0; j < 8; ++j) {
      OUT[(size_t)(out_row0 + j + 8 * half) * OUT_DIM + n] = acc[t][j] + bv;
    }
  }
}

// ---------------------------------------------------------------------------
// Host launcher. Input dict order:
// 0 node_feat, 1 edge_feat, 2 Wn, 3 bn, 4 We, 5 be, 6 Wc, 7 bc, 8 src, 9 dst
// ---------------------------------------------------------------------------
extern "C" void kernel_launch(void* const* d_in, const int* in_sizes, int n_in,
                              void* d_out, int out_size, void* d_ws, size_t ws_size,
                              hipStream_t stream) {
  const float* node_feat = (const float*)d_in[0];
  const float* edge_feat = (const float*)d_in[1];
  const float* Wn = (const float*)d_in[2];
  const float* bn = (const float*)d_in[3];
  const float* We = (const float*)d_in[4];
  const float* be = (const float*)d_in[5];
  const float* Wc = (const float*)d_in[6];
  const float* bc = (const float*)d_in[7];
  const int*   src = (const int*)d_in[8];
  const int*   dst = (const int*)d_in[9];
  float* out = (float*)d_out;

  // Workspace layout: h [N,128] | agg_sum [N,128] | deg [N] | WTn | WTe | WTc
  float* h   = (float*)d_ws;
  float* agg = h   + (size_t)N_NODES * OUT_DIM;
  float* deg = agg + (size_t)N_NODES * OUT_DIM;
  float* wtn = deg + N_NODES;                       // [128][256]
  float* wte = wtn + NODE_DIM * OUT_DIM;            // [128][128]
  float* wtc = wte + EDGE_DIM * OUT_DIM;            // [128][256]

  // Zero accumulation buffers (agg and deg are contiguous).
  hipMemsetAsync(agg, 0,
                 ((size_t)N_NODES * OUT_DIM + N_NODES) * sizeof(float), stream);

  // 0) transpose weights (tiny; makes every B fragment a single b64 load)
  transpose_kernel<<<(NODE_DIM * OUT_DIM + 255) / 256, 256, 0, stream>>>(
      Wn, wtn, NODE_DIM, OUT_DIM);
  transpose_kernel<<<(EDGE_DIM * OUT_DIM + 255) / 256, 256, 0, stream>>>(
      We, wte, EDGE_DIM, OUT_DIM);
  transpose_kernel<<<(2 * OUT_DIM * OUT_DIM + 255) / 256, 256, 0, stream>>>(
      Wc, wtc, 2 * OUT_DIM, OUT_DIM);

  // 1) h = node_feat @ Wn + bn     (2500 wave-tiles / 4 waves per block)
  gemm_h_kernel<<<(N_NODES / 16) / WAVES_PER_BLOCK, 128, 0, stream>>>(
      node_feat, wtn, bn, h);

  // 2) degree counts
  deg_kernel<<<(N_EDGES + 255) / 256, 256, 0, stream>>>(dst, deg);

  // 3) fused edge GEMM + gather(h[src]) + scatter-add into agg_sum
  edge_gemm_scatter_kernel<<<(N_EDGES / 16) / WAVES_PER_BLOCK, 128, 0, stream>>>(
      edge_feat, wte, be, h, src, dst, agg);

  // 4) out = concat(h, agg/max(deg,1)) @ Wc + bc
  combine_kernel<<<(N_NODES / 16) / WAVES_PER_BLOCK, 128, 0, stream>>>(
      h, agg, deg, wtc, bc, out);
}